// GNN_47914655154384
// MI455X (gfx1250) — compile-verified
//
#include <hip/hip_runtime.h>
#include <float.h>

// ---------------------------------------------------------------------------
// 2-layer GAT on MI455X (gfx1250, wave32).
//   layer: h = in @ W^T + b           (WMMA f32 16x16x4 GEMM)
//          a_i[n] = att[0:64]  . h[n] ; a_j[n] = att[64:128] . h[n]
//          score_e = leaky_relu(a_i[dst] + a_j[src])
//          softmax grouped by src (atomic max + atomic add)
//          aggr[dst] += alpha_e * h[src]  (16 lanes/edge, float4 + f32 atomics)
//          out = relu(aggr + bias)
//   final: y[n] = n2[n] . Wout + bout   (wave32 shuffle reduction)
// All working arrays (~43 MB) live in d_ws and fit in the 192 MB L2.
// ---------------------------------------------------------------------------

typedef __attribute__((ext_vector_type(2))) float v2f;
typedef __attribute__((ext_vector_type(8))) float v8f;

#define NEG_SLOPE 0.01f

// ---------------------------------------------------------------------------
// init per-layer state: m = -FLT_MAX (bit pattern), denom = 0, aggr = 0
// ---------------------------------------------------------------------------
__global__ void k_init(int* __restrict__ m_bits, float* __restrict__ denom,
                       float* __restrict__ aggr, int Nn) {
  int i = blockIdx.x * blockDim.x + threadIdx.x;
  if (i < Nn) { m_bits[i] = 0xFF7FFFFF; denom[i] = 0.0f; }  // -FLT_MAX bits
  if (i < Nn * 64) aggr[i] = 0.0f;
}

// ---------------------------------------------------------------------------
// WMMA GEMM: h[M x 64] = in[M x K] @ W[64 x K]^T + b.   M % 16 == 0.
// One wave computes one 16-row strip (4 accumulators of 16x16).
// f32 WMMA 16x16x4 layouts (ISA 7.12.2):
//   A 16x4 : lane l<16 holds row M=l, K = {0,1}; lanes 16-31 hold K = {2,3}
//   B 4x16 : VGPR0: K=0(lanes0-15)/K=2(lanes16-31); VGPR1: K=1/K=3; N = lane&15
//   C 16x16: VGPR v: M = v + (lane>=16 ? 8 : 0), N = lane&15
// ---------------------------------------------------------------------------
template <int K>
__global__ void k_gemm_wmma(const float* __restrict__ in, const float* __restrict__ W,
                            const float* __restrict__ b, float* __restrict__ h, int M) {
  int wave = (int)((blockIdx.x * blockDim.x + threadIdx.x) >> 5);
  int lane = threadIdx.x & 31;
  int row0 = wave * 16;
  if (row0 >= M) return;                 // wave-uniform: EXEC stays all-ones
  int half = lane >> 4;                  // 0: K-pair {0,1}, 1: K-pair {2,3}
  int l16  = lane & 15;

  v8f acc0 = {}, acc1 = {}, acc2 = {}, acc3 = {};
  const float* arow = in + (size_t)(row0 + l16) * K;

  #pragma unroll
  for (int k0 = 0; k0 < K; k0 += 4) {
    int kb = k0 + half * 2;
    v2f a;  a.x = arow[kb];  a.y = arow[kb + 1];
    // B[k][n] = W[n][k] ; n = t*16 + l16
    v2f b0, b1, b2, b3;
    { const float* p = W + (size_t)(0  + l16) * K + kb; b0.x = p[0]; b0.y = p[1]; }
    { const float* p = W + (size_t)(16 + l16) * K + kb; b1.x = p[0]; b1.y = p[1]; }
    { const float* p = W + (size_t)(32 + l16) * K + kb; b2.x = p[0]; b2.y = p[1]; }
    { const float* p = W + (size_t)(48 + l16) * K + kb; b3.x = p[0]; b3.y = p[1]; }
    acc0 = __builtin_amdgcn_wmma_f32_16x16x4_f32(false, a, false, b0, (short)0, acc0, false, false);
    acc1 = __builtin_amdgcn_wmma_f32_16x16x4_f32(false, a, false, b1, (short)0, acc1, false, false);
    acc2 = __builtin_amdgcn_wmma_f32_16x16x4_f32(false, a, false, b2, (short)0, acc2, false, false);
    acc3 = __builtin_amdgcn_wmma_f32_16x16x4_f32(false, a, false, b3, (short)0, acc3, false, false);
  }

  // store D + bias
  #pragma unroll
  for (int v = 0; v < 8; ++v) {
    int m = row0 + v + half * 8;
    float* hrow = h + (size_t)m * 64 + l16;
    hrow[0]  = acc0[v] + b[0  + l16];
    hrow[16] = acc1[v] + b[16 + l16];
    hrow[32] = acc2[v] + b[32 + l16];
    hrow[48] = acc3[v] + b[48 + l16];
  }
}

// ---------------------------------------------------------------------------
// per-node attention scalars: a_i = att[0:64].h[n], a_j = att[64:128].h[n]
// one wave32 per node, shuffle reduction
// ---------------------------------------------------------------------------
__global__ void k_node_att(const float* __restrict__ h, const float* __restrict__ att,
                           float* __restrict__ a_i, float* __restrict__ a_j, int Nn) {
  int n = blockIdx.x * (blockDim.x >> 5) + (threadIdx.x >> 5);
  int lane = threadIdx.x & 31;
  if (n >= Nn) return;
  const float* hp = h + (size_t)n * 64;
  float h0 = hp[lane], h1 = hp[lane + 32];
  float vi = h0 * att[lane]      + h1 * att[lane + 32];
  float vj = h0 * att[64 + lane] + h1 * att[96 + lane];
  #pragma unroll
  for (int off = 16; off > 0; off >>= 1) {
    vi += __shfl_xor(vi, off, 32);
    vj += __shfl_xor(vj, off, 32);
  }
  if (lane == 0) { a_i[n] = vi; a_j[n] = vj; }
}

__device__ __forceinline__ void get_edge(const int* __restrict__ ei, int e, int E,
                                         int& s, int& d) {
  if (e < E) { s = ei[e]; d = ei[E + e]; }
  else       { s = e - E; d = e - E; }      // self loops appended
}

// ---------------------------------------------------------------------------
// pass 1: score + segment max over src (mixed-sign float atomic-max trick)
// ---------------------------------------------------------------------------
__global__ void k_edge_score(const int* __restrict__ ei, const float* __restrict__ a_i,
                             const float* __restrict__ a_j, float* __restrict__ score,
                             int* __restrict__ m_bits, int E, int Etot) {
  int e = blockIdx.x * blockDim.x + threadIdx.x;
  if (e >= Etot) return;
  int s, d; get_edge(ei, e, E, s, d);
  float sc = a_i[d] + a_j[s];
  sc = sc > 0.0f ? sc : sc * NEG_SLOPE;     // leaky_relu
  score[e] = sc;
  if (sc >= 0.0f) atomicMax(&m_bits[s], __float_as_int(sc));
  else            atomicMin((unsigned int*)&m_bits[s], __float_as_uint(sc));
}

// ---------------------------------------------------------------------------
// pass 2: exp(score - m[src]) (in place) + segment sum over src
// ---------------------------------------------------------------------------
__global__ void k_edge_expsum(const int* __restrict__ ei, const float* __restrict__ m,
                              float* __restrict__ score, float* __restrict__ denom,
                              int E, int Etot) {
  int e = blockIdx.x * blockDim.x + threadIdx.x;
  if (e >= Etot) return;
  int s, d; get_edge(ei, e, E, s, d);
  float ev = __expf(score[e] - m[s]);
  score[e] = ev;
  atomicAdd(&denom[s], ev);
}

// ---------------------------------------------------------------------------
// pass 3: aggr[dst] += alpha * h[src].  16 threads per edge, 4 cols each.
// Consecutive lanes -> consecutive columns: coalesced float4 gather and
// contiguous atomic adds (L2-friendly).
// ---------------------------------------------------------------------------
__global__ void k_edge_message(const int* __restrict__ ei, const float* __restrict__ escore,
                               const float* __restrict__ denom, const float* __restrict__ h,
                               float* __restrict__ aggr, int E, int Etot) {
  long long t = (long long)blockIdx.x * blockDim.x + threadIdx.x;
  int e  = (int)(t >> 4);
  int c4 = ((int)t & 15) * 4;
  if (e >= Etot) return;
  int s, d; get_edge(ei, e, E, s, d);
  float alpha = escore[e] / (denom[s] + 1e-16f);
  const float4 hv = *(const float4*)(h + (size_t)s * 64 + c4);
  float* ap = aggr + (size_t)d * 64 + c4;
  atomicAdd(ap + 0, alpha * hv.x);
  atomicAdd(ap + 1, alpha * hv.y);
  atomicAdd(ap + 2, alpha * hv.z);
  atomicAdd(ap + 3, alpha * hv.w);
}

// ---------------------------------------------------------------------------
// out = relu(aggr + bias)   (relu(relu(x)) == relu(x))
// ---------------------------------------------------------------------------
__global__ void k_node_finish(const float* __restrict__ aggr, const float* __restrict__ bias,
                              float* __restrict__ out, int Nn) {
  int i = blockIdx.x * blockDim.x + threadIdx.x;
  if (i >= Nn * 64) return;
  float v = aggr[i] + bias[i & 63];
  out[i] = v > 0.0f ? v : 0.0f;
}

// ---------------------------------------------------------------------------
// y[n] = n2[n] . Wout + bout  — one wave32 per node
// ---------------------------------------------------------------------------
__global__ void k_out_proj(const float* __restrict__ n2, const float* __restrict__ Wout,
                           const float* __restrict__ bout, float* __restrict__ y, int Nn) {
  int n = blockIdx.x * (blockDim.x >> 5) + (threadIdx.x >> 5);
  int lane = threadIdx.x & 31;
  if (n >= Nn) return;
  const float* hp = n2 + (size_t)n * 64;
  float v = hp[lane] * Wout[lane] + hp[lane + 32] * Wout[lane + 32];
  #pragma unroll
  for (int off = 16; off > 0; off >>= 1) v += __shfl_xor(v, off, 32);
  if (lane == 0) y[n] = v + bout[0];
}

// ---------------------------------------------------------------------------
extern "C" void kernel_launch(void* const* d_in, const int* in_sizes, int n_in,
                              void* d_out, int out_size, void* d_ws, size_t ws_size,
                              hipStream_t stream) {
  const float* x     = (const float*)d_in[0];
  const int*   ei    = (const int*)  d_in[1];   // [2, E] int32
  const float* W1    = (const float*)d_in[2];
  const float* b1    = (const float*)d_in[3];
  const float* att1  = (const float*)d_in[4];
  const float* bias1 = (const float*)d_in[5];
  const float* W2    = (const float*)d_in[6];
  const float* b2    = (const float*)d_in[7];
  const float* att2  = (const float*)d_in[8];
  const float* bias2 = (const float*)d_in[9];
  const float* Wout  = (const float*)d_in[10];
  const float* bout  = (const float*)d_in[11];
  float* y = (float*)d_out;

  const int Nn   = in_sizes[0] / 16;   // 50000
  const int E    = in_sizes[1] / 2;    // 800000
  const int Etot = E + Nn;             // with self loops

  // ---- workspace layout (256B aligned) ----
  char* p = (char*)d_ws;
  auto alloc = [&](size_t bytes) {
    char* r = p;
    p += (bytes + 255) & ~(size_t)255;
    return r;
  };
  float* h      = (float*)alloc((size_t)Nn * 64 * sizeof(float));  // linear output
  float* nbuf   = (float*)alloc((size_t)Nn * 64 * sizeof(float));  // activated nodes
  float* aggr   = (float*)alloc((size_t)Nn * 64 * sizeof(float));
  float* score  = (float*)alloc((size_t)Etot * sizeof(float));
  float* a_i    = (float*)alloc((size_t)Nn * sizeof(float));
  float* a_j    = (float*)alloc((size_t)Nn * sizeof(float));
  int*   m_bits = (int*)  alloc((size_t)Nn * sizeof(float));
  float* denom  = (float*)alloc((size_t)Nn * sizeof(float));

  const int TB = 256;
  dim3 blk(TB);
  int g_init  = (Nn * 64 + TB - 1) / TB;
  int g_gemm  = ((Nn / 16) + 7) / 8;          // 8 waves per block, 1 row-tile each
  int g_node  = (Nn + 7) / 8;                 // 8 waves per block, 1 node each
  int g_edge  = (Etot + TB - 1) / TB;
  int g_msg   = (int)(((long long)Etot * 16 + TB - 1) / TB);
  int g_fin   = (Nn * 64 + TB - 1) / TB;

  // -------- layer 1 (K = 16) --------
  k_init<<<g_init, blk, 0, stream>>>(m_bits, denom, aggr, Nn);
  k_gemm_wmma<16><<<g_gemm, blk, 0, stream>>>(x, W1, b1, h, Nn);
  k_node_att<<<g_node, blk, 0, stream>>>(h, att1, a_i, a_j, Nn);
  k_edge_score<<<g_edge, blk, 0, stream>>>(ei, a_i, a_j, score, m_bits, E, Etot);
  k_edge_expsum<<<g_edge, blk, 0, stream>>>(ei, (const float*)m_bits, score, denom, E, Etot);
  k_edge_message<<<g_msg, blk, 0, stream>>>(ei, score, denom, h, aggr, E, Etot);
  k_node_finish<<<g_fin, blk, 0, stream>>>(aggr, bias1, nbuf, Nn);

  // -------- layer 2 (K = 64) --------
  k_init<<<g_init, blk, 0, stream>>>(m_bits, denom, aggr, Nn);
  k_gemm_wmma<64><<<g_gemm, blk, 0, stream>>>(nbuf, W2, b2, h, Nn);
  k_node_att<<<g_node, blk, 0, stream>>>(h, att2, a_i, a_j, Nn);
  k_edge_score<<<g_edge, blk, 0, stream>>>(ei, a_i, a_j, score, m_bits, E, Etot);
  k_edge_expsum<<<g_edge, blk, 0, stream>>>(ei, (const float*)m_bits, score, denom, E, Etot);
  k_edge_message<<<g_msg, blk, 0, stream>>>(ei, score, denom, h, aggr, E, Etot);
  k_node_finish<<<g_fin, blk, 0, stream>>>(aggr, bias2, nbuf, Nn);

  // -------- output projection --------
  k_out_proj<<<g_node, blk, 0, stream>>>(nbuf, Wout, bout, y, Nn);
}